// SonicCruncher_24670292148708
// MI455X (gfx1250) — compile-verified
//
#include <hip/hip_runtime.h>
#include <math.h>

#define DIM 512
#define NROWS 8192
#define MAXK 4
#define WAVES 8
#define TROWS 16

typedef __attribute__((ext_vector_type(2))) float v2f;
typedef __attribute__((ext_vector_type(4))) float v4f;
typedef __attribute__((ext_vector_type(8))) float v8f;

// ---------------- Phase A: qk_pad[16][512] = scale * (nexus[:k] @ Wq^T) @ Wk ----------------
__global__ void qk_precompute(const float* __restrict__ nexus,
                              const int* __restrict__ kptr,
                              const float* __restrict__ Wq,
                              const float* __restrict__ Wk,
                              float* __restrict__ qkpad) {
  __shared__ float qlds[MAXK * DIM];
  const int tid = threadIdx.x;
  int krt = *kptr; if (krt < 0) krt = 0; if (krt > MAXK) krt = MAXK;
  for (int e = tid; e < DIM; e += 256) {
    float acc[MAXK] = {0.f, 0.f, 0.f, 0.f};
    for (int d = 0; d < DIM; ++d) {
      float w = Wq[e * DIM + d];
#pragma unroll
      for (int kk = 0; kk < MAXK; ++kk) acc[kk] += nexus[kk * DIM + d] * w;
    }
#pragma unroll
    for (int kk = 0; kk < MAXK; ++kk) qlds[kk * DIM + e] = (kk < krt) ? acc[kk] : 0.f;
  }
  __syncthreads();
  const float scale = 0.04419417382415922f; // 1/sqrt(512)
  for (int d = tid; d < DIM; d += 256) {
    float acc[MAXK] = {0.f, 0.f, 0.f, 0.f};
    for (int e = 0; e < DIM; ++e) {
      float w = Wk[e * DIM + d];
#pragma unroll
      for (int kk = 0; kk < MAXK; ++kk) acc[kk] += qlds[kk * DIM + e] * w;
    }
#pragma unroll
    for (int kk = 0; kk < MAXK; ++kk) qkpad[kk * DIM + d] = acc[kk] * scale;
    for (int kk = MAXK; kk < 16; ++kk) qkpad[kk * DIM + d] = 0.f; // dead WMMA columns
  }
}

// ---------------- Main streaming flash kernel ----------------
// grid = BH*nchunk blocks x 256 threads. Each wave owns a 16-row tile loop with
// a private 32KB swizzled LDS slab + 8KB LDS accumulator. One HBM pass total.
__global__ void flash_stream(const float* __restrict__ ckv,
                             const float* __restrict__ qkpad,
                             float* __restrict__ pm, float* __restrict__ pl,
                             float* __restrict__ pacc,
                             int nchunk, int rows_per_chunk) {
  extern __shared__ float smem[];
  float* tiles = smem;                         // [WAVES][16][512], XOR-swizzled
  float* accs  = smem + WAVES * TROWS * DIM;   // [WAVES][MAXK][512]
  const int tid  = threadIdx.x;
  const int wave = tid >> 5;
  const int lane = tid & 31;
  const int bh    = blockIdx.x / nchunk;
  const int chunk = blockIdx.x % nchunk;
  const float* base = ckv + ((size_t)bh * NROWS + (size_t)chunk * rows_per_chunk) * DIM;
  float* mytile = tiles + wave * (TROWS * DIM);
  float* myacc  = accs  + wave * (MAXK * DIM);

  for (int i = lane; i < MAXK * DIM; i += 32) myacc[i] = 0.f;

  const int r   = lane & 15;      // A-matrix row / D-matrix column index
  const int hi  = lane >> 4;      // half-wave
  const int off = hi << 1;        // K-pair offset (0 or 2)
  const int rx  = r << 2;
  const int nhi = hi << 3;

  float m_run = -3.0e38f, l_run = 0.f;
  const int ntiles = rows_per_chunk / TROWS;

  for (int t = wave; t < ntiles; t += WAVES) {
    const float* rowbase = base + (size_t)t * TROWS * DIM;
    if (t + WAVES < ntiles)
      __builtin_prefetch(base + (size_t)(t + WAVES) * TROWS * DIM, 0, 0);

    // ---- stage 16x512 f32 contiguously -> LDS, col ^ (4*row) swizzle ----
#pragma unroll 4
    for (int i = 0; i < 64; ++i) {
      int f  = i * 32 + lane;          // float4 index 0..2047 (contiguous in HBM)
      int rr = f >> 7;
      int c  = (f & 127) << 2;
      v4f v = *(const v4f*)(rowbase + ((size_t)f << 2));
      *(v4f*)(mytile + rr * DIM + (c ^ (rr << 2))) = v;
    }

    // ---- logits: D(16n x 16k) += A(tile 16x4) * B(qk^T 4x16), 128 steps of K=4 ----
    v8f dfrag = {0.f, 0.f, 0.f, 0.f, 0.f, 0.f, 0.f, 0.f};
    const float* qrow = qkpad + r * DIM + off;
    const float* arow = mytile + r * DIM;
#pragma unroll 4
    for (int j = 0; j < 128; ++j) {
      v2f a = *(const v2f*)(arow + (((j << 2) ^ rx) + off)); // tile[r][4j+off..+1]
      v2f b = *(const v2f*)(qrow + (j << 2));                // qk[r][4j+off..+1]
      dfrag = __builtin_amdgcn_wmma_f32_16x16x4_f32(false, a, false, b,
                                                    (short)0, dfrag, false, false);
    }

    // ---- online softmax (lane's column k = r; rows split across lane pair) ----
    float tmax = dfrag[0];
#pragma unroll
    for (int i2 = 1; i2 < 8; ++i2) tmax = fmaxf(tmax, dfrag[i2]);
    tmax = fmaxf(tmax, __shfl_xor(tmax, 16, 32));
    float mnew = fmaxf(m_run, tmax);
    float p[8];
    float tsum = 0.f;
#pragma unroll
    for (int i2 = 0; i2 < 8; ++i2) { p[i2] = __expf(dfrag[i2] - mnew); tsum += p[i2]; }
    tsum += __shfl_xor(tsum, 16, 32);
    float corr = __expf(m_run - mnew);
    l_run = l_run * corr + tsum;
    m_run = mnew;
    float cr0 = __shfl(corr, 0, 32);
    float cr1 = __shfl(corr, 1, 32);
    float cr2 = __shfl(corr, 2, 32);
    float cr3 = __shfl(corr, 3, 32);

    // ---- accumulation: acc[k,d] += sum_n p[k,n]*tile[n,d]; probs fragment IS the A operand ----
#pragma unroll 2
    for (int c = 0; c < 32; ++c) {
      int dl = (c << 4) + r;               // this lane's d column
      v8f cfrag = {0.f, 0.f, 0.f, 0.f, 0.f, 0.f, 0.f, 0.f};
      if (lane < 16) {                     // rows k=0..3 live in VGPR 0..3 of lanes 0-15
        cfrag[0] = myacc[0 * DIM + dl] * cr0;
        cfrag[1] = myacc[1 * DIM + dl] * cr1;
        cfrag[2] = myacc[2 * DIM + dl] * cr2;
        cfrag[3] = myacc[3 * DIM + dl] * cr3;
      }
#pragma unroll
      for (int q = 0; q < 4; ++q) {        // n-quad {q, q+4, q+8, q+12}
        int n0 = q + nhi;
        int n1 = q + 4 + nhi;
        v2f a = {p[q], p[q + 4]};
        v2f b = {mytile[n0 * DIM + (dl ^ (n0 << 2))],
                 mytile[n1 * DIM + (dl ^ (n1 << 2))]};
        cfrag = __builtin_amdgcn_wmma_f32_16x16x4_f32(false, a, false, b,
                                                      (short)0, cfrag, false, false);
      }
      if (lane < 16) {
        myacc[0 * DIM + dl] = cfrag[0];
        myacc[1 * DIM + dl] = cfrag[1];
        myacc[2 * DIM + dl] = cfrag[2];
        myacc[3 * DIM + dl] = cfrag[3];
      }
    }
  }

  // publish per-wave (m,l) in the now-free tile slab, then block-combine
  if (lane < MAXK) { mytile[lane] = m_run; mytile[MAXK + lane] = l_run; }
  __syncthreads();

  const int blk = blockIdx.x;
  for (int g = tid; g < MAXK * DIM; g += 256) {
    int kk = g >> 9;
    int dd = g & (DIM - 1);
    float M = -3.0e38f;
#pragma unroll
    for (int w = 0; w < WAVES; ++w) M = fmaxf(M, tiles[w * (TROWS * DIM) + kk]);
    float L = 0.f, S = 0.f;
#pragma unroll
    for (int w = 0; w < WAVES; ++w) {
      float e = __expf(tiles[w * (TROWS * DIM) + kk] - M);
      L += e * tiles[w * (TROWS * DIM) + MAXK + kk];
      S += e * accs[w * (MAXK * DIM) + kk * DIM + dd];
    }
    pacc[(size_t)blk * (MAXK * DIM) + g] = S;
    if (dd == 0) { pm[blk * MAXK + kk] = M; pl[blk * MAXK + kk] = L; }
  }
}

// ---------------- Finalize: cross-chunk combine, then Wv and Wo projections ----------------
__global__ void finalize(const float* __restrict__ pm, const float* __restrict__ pl,
                         const float* __restrict__ pacc,
                         const float* __restrict__ Wv, const float* __restrict__ Wo,
                         const int* __restrict__ kptr,
                         float* __restrict__ out, int nchunk) {
  __shared__ float s[MAXK * DIM];
  __shared__ float cn[MAXK * DIM];
  const int tid = threadIdx.x;
  const int bh  = blockIdx.x;
  int krt = *kptr; if (krt < 1) krt = 1; if (krt > MAXK) krt = MAXK;

  for (int g = tid; g < MAXK * DIM; g += 256) {
    int kk = g >> 9;
    int dd = g & (DIM - 1);
    float M = -3.0e38f;
    for (int c = 0; c < nchunk; ++c) M = fmaxf(M, pm[(bh * nchunk + c) * MAXK + kk]);
    float L = 0.f, S = 0.f;
    for (int c = 0; c < nchunk; ++c) {
      int pi = bh * nchunk + c;
      float e = __expf(pm[pi * MAXK + kk] - M);
      L += e * pl[pi * MAXK + kk];
      S += e * pacc[(size_t)pi * (MAXK * DIM) + kk * DIM + dd];
    }
    s[g] = S / L;   // normalized prob-weighted sum of raw c_kv rows
  }
  __syncthreads();
  for (int e = tid; e < DIM; e += 256) {
    float acc[MAXK] = {0.f, 0.f, 0.f, 0.f};
    for (int d = 0; d < DIM; ++d) {
      float w = Wv[e * DIM + d];
#pragma unroll
      for (int kk = 0; kk < MAXK; ++kk) acc[kk] += s[kk * DIM + d] * w;
    }
#pragma unroll
    for (int kk = 0; kk < MAXK; ++kk) cn[kk * DIM + e] = acc[kk];
  }
  __syncthreads();
  for (int e = tid; e < DIM; e += 256) {
    float acc[MAXK] = {0.f, 0.f, 0.f, 0.f};
    for (int d = 0; d < DIM; ++d) {
      float w = Wo[e * DIM + d];
#pragma unroll
      for (int kk = 0; kk < MAXK; ++kk) acc[kk] += cn[kk * DIM + d] * w;
    }
    for (int kk = 0; kk < krt; ++kk)
      out[((size_t)bh * krt + kk) * DIM + e] = acc[kk];
  }
}

extern "C" void kernel_launch(void* const* d_in, const int* in_sizes, int n_in,
                              void* d_out, int out_size, void* d_ws, size_t ws_size,
                              hipStream_t stream) {
  const float* ckv   = (const float*)d_in[0];
  const int*   kptr  = (const int*)d_in[1];
  const float* nexus = (const float*)d_in[2];
  const float* Wq    = (const float*)d_in[3];
  const float* Wk    = (const float*)d_in[4];
  const float* Wv    = (const float*)d_in[5];
  const float* Wo    = (const float*)d_in[6];
  float* out = (float*)d_out;

  long long total = in_sizes[0];
  int BH = (int)(total / ((long long)NROWS * DIM));   // 2*16 = 32

  // choose chunk count to fit workspace (partial = 4 m + 4 l + 4*512 acc floats)
  int nchunk = 16;
  while (nchunk > 1) {
    size_t need = 32768 + (size_t)BH * nchunk * (MAXK * DIM + 2 * MAXK) * sizeof(float) + 256;
    if (need <= ws_size) break;
    nchunk >>= 1;
  }
  int rows_per_chunk = NROWS / nchunk;

  float* qkpad = (float*)d_ws;
  float* pm    = (float*)((char*)d_ws + 32768);
  float* pl    = pm + (size_t)BH * nchunk * MAXK;
  float* pacc  = pl + (size_t)BH * nchunk * MAXK;

  qk_precompute<<<1, 256, 0, stream>>>(nexus, kptr, Wq, Wk, qkpad);

  size_t smem = (size_t)(WAVES * TROWS * DIM + WAVES * MAXK * DIM) * sizeof(float); // 320KB
  flash_stream<<<BH * nchunk, 256, smem, stream>>>(ckv, qkpad, pm, pl, pacc,
                                                   nchunk, rows_per_chunk);

  finalize<<<BH, 256, 0, stream>>>(pm, pl, pacc, Wv, Wo, kptr, out, nchunk);
}